// DataAugmenter_4303557231052
// MI455X (gfx1250) — compile-verified
//
#include <hip/hip_runtime.h>
#include <stdint.h>

#define HN 128
#define HC 63.5f

typedef uint32_t v4u __attribute__((ext_vector_type(4)));
typedef uint32_t v8u __attribute__((ext_vector_type(8)));

// ---------------- threefry2x32 (JAX-flavored counter RNG, deterministic) ----------------
struct U2 { uint32_t x, y; };

__device__ __forceinline__ uint32_t rotl32(uint32_t x, int r) { return (x << r) | (x >> (32 - r)); }

__device__ __forceinline__ U2 threefry2x32(uint32_t k0, uint32_t k1, uint32_t c0, uint32_t c1) {
    uint32_t ks0 = k0, ks1 = k1, ks2 = 0x1BD11BDAu ^ k0 ^ k1;
    uint32_t x0 = c0 + ks0, x1 = c1 + ks1;
    const int R[8] = {13, 15, 26, 6, 17, 29, 16, 24};
#pragma unroll
    for (int g = 0; g < 5; ++g) {
#pragma unroll
        for (int j = 0; j < 4; ++j) {
            x0 += x1;
            x1 = rotl32(x1, R[(g & 1) * 4 + j]);
            x1 ^= x0;
        }
        uint32_t inj = (uint32_t)(g + 1);
        uint32_t kk0 = (inj % 3 == 0) ? ks0 : (inj % 3 == 1) ? ks1 : ks2;
        uint32_t m = (inj + 1) % 3;
        uint32_t kk1 = (m == 0) ? ks0 : (m == 1) ? ks1 : ks2;
        x0 += kk0;
        x1 += kk1 + inj;
    }
    U2 r; r.x = x0; r.y = x1; return r;
}

__device__ __forceinline__ float u01(uint32_t b) { return (float)(b >> 8) * (1.0f / 16777216.0f); }

// ---------------- kernel 1: per-sample augmentation parameters + min/max init ----------------
__global__ void aug_params_kernel(float* __restrict__ params, float* __restrict__ minmax, int B) {
    int b = (int)threadIdx.x;
    if (b >= B) return;
    U2 kb = threefry2x32(0u, 42u, 2u * (uint32_t)b, 2u * (uint32_t)b + 1u);
    U2 kd = threefry2x32(kb.x, kb.y, 0u, 1u);
    U2 kz = threefry2x32(kb.x, kb.y, 2u, 3u);
    U2 ks = threefry2x32(kb.x, kb.y, 4u, 5u);
    U2 kn = threefry2x32(kb.x, kb.y, 6u, 7u);
    U2 kg = threefry2x32(kb.x, kb.y, 8u, 9u);
    float u[6];
#pragma unroll
    for (int i = 0; i < 6; ++i) u[i] = u01(threefry2x32(kd.x, kd.y, (uint32_t)i, 0x2545F491u).x);
    float z     = (u[0] < 0.15f) ? (0.7f + 0.3f * u01(threefry2x32(kz.x, kz.y, 0u, 1u).x)) : 1.0f;
    float stdev = (u[4] < 0.10f) ? (0.2f * u01(threefry2x32(ks.x, ks.y, 0u, 1u).x)) : 0.0f;
    float gamma = (u[5] < 0.10f) ? (0.8f + 0.4f * u01(threefry2x32(kg.x, kg.y, 0u, 1u).x)) : 1.0f;
    float* P = params + b * 8;
    P[0] = z;
    P[1] = (u[1] < 0.5f) ? 1.0f : 0.0f;   // flip H
    P[2] = (u[2] < 0.5f) ? 1.0f : 0.0f;   // flip W
    P[3] = (u[3] < 0.5f) ? 1.0f : 0.0f;   // flip D
    P[4] = stdev;
    P[5] = gamma;
    P[6] = __uint_as_float(kn.x);
    P[7] = __uint_as_float(kn.y);
    minmax[2 * b]     =  INFINITY;   // re-initialized every call (deterministic)
    minmax[2 * b + 1] = -INFINITY;
}

// ---------------- kernel 2: one block per output line (b,h,w,:) ----------------
// TDM stages the 2x2x128 image tile; async-LDS stages the label row.
__global__ __launch_bounds__(128) void aug_line_kernel(
    const float* __restrict__ img, const float* __restrict__ lbl,
    float* __restrict__ oimg, float* __restrict__ olbl,
    const float* __restrict__ params, float* __restrict__ minmax) {
    __shared__ __align__(16) float rows[4][HN];   // TDM tile: (ho,wo),(ho,wo+1),(ho+1,wo),(ho+1,wo+1)
    __shared__ __align__(16) float lrow[HN];      // nearest label row
    __shared__ float smin[128], smax[128];

    const uint32_t lin = blockIdx.x;
    const int b = (int)(lin >> 14);           // 128*128 = 2^14 lines per sample
    const int h = (int)((lin >> 7) & 127u);
    const int w = (int)(lin & 127u);

    const float* P = params + b * 8;
    const float z = P[0];
    const bool fh = (P[1] != 0.0f), fw = (P[2] != 0.0f), fd = (P[3] != 0.0f);
    const float stdev = P[4];
    const uint32_t kn0 = __float_as_uint(P[6]);
    const uint32_t kn1 = __float_as_uint(P[7]);
    const float invz = 1.0f / z;

    // H / W axes: block-uniform interpolation setup (flip folded into index map)
    const int eh = fh ? (HN - 1 - h) : h;
    const int ew = fw ? (HN - 1 - w) : w;
    const float sh = (eh - HC) * invz + HC;
    const float sw = (ew - HC) * invz + HC;
    const int h0i = (int)floorf(sh), w0i = (int)floorf(sw);
    const float th = sh - (float)h0i, tw = sw - (float)w0i;
    const float wh0 = (1.0f - th) * ((h0i     >= 0 && h0i     < HN) ? 1.0f : 0.0f);
    const float wh1 = th          * ((h0i + 1 >= 0 && h0i + 1 < HN) ? 1.0f : 0.0f);
    const float ww0 = (1.0f - tw) * ((w0i     >= 0 && w0i     < HN) ? 1.0f : 0.0f);
    const float ww1 = tw          * ((w0i + 1 >= 0 && w0i + 1 < HN) ? 1.0f : 0.0f);

    // TDM tile origin (always fully in-bounds); map interp weights onto the 2 loaded slots
    const int ho = min(max(h0i, 0), HN - 2);
    const int wo = min(max(w0i, 0), HN - 2);
    float wsh[2], wsw[2];
#pragma unroll
    for (int k = 0; k < 2; ++k) {
        wsh[k] = (((ho + k) == h0i) ? wh0 : 0.0f) + (((ho + k) == (h0i + 1)) ? wh1 : 0.0f);
        wsw[k] = (((wo + k) == w0i) ? ww0 : 0.0f) + (((wo + k) == (w0i + 1)) ? ww1 : 0.0f);
    }

    // nearest (label) indices for H/W
    const int hni = (int)rintf(sh), wni = (int)rintf(sw);
    const bool hv = (hni >= 0) && (hni < HN), wv = (wni >= 0) && (wni < HN);
    const int hn = min(max(hni, 0), HN - 1), wn = min(max(wni, 0), HN - 1);

    const size_t sampOff = (size_t)b * (size_t)(HN * HN * HN);
    const float* ib = img + sampOff;
    const float* lb = lbl + sampOff;

    // ---- wave 0: TDM load of the 2x2x128 image tile + async-LDS load of label row ----
    if (threadIdx.x < 32u) {
        // Tensor DMA descriptor (ISA ch.8): 3D tile, dim0=d (contig), dim1=w, dim2=h
        const uint64_t ga = (uint64_t)(uintptr_t)(ib + ((size_t)ho * HN + wo) * HN);
        const uint32_t ldsBase = (uint32_t)(uintptr_t)(&rows[0][0]);
        v4u g0;
        g0[0] = 1u;                                                  // count=1 (valid), no gather
        g0[1] = ldsBase;                                             // lds_addr
        g0[2] = (uint32_t)ga;                                        // global_addr[31:0]
        g0[3] = ((uint32_t)(ga >> 32) & 0x01FFFFFFu) | (2u << 30);   // global_addr[56:32] | type=2
        v8u g1;
        g1[0] = 2u << 16;                 // workgroup_mask=0, data_size=2 (4B)
        g1[1] = (uint32_t)HN << 16;       // tensor_dim0[15:0]=128 (abar addr=0)
        g1[2] = (uint32_t)HN << 16;       // tensor_dim0[31:16]=0, tensor_dim1[15:0]=128
        g1[3] = (uint32_t)HN << 16;       // tensor_dim1[31:16]=0, tile_dim0=128
        g1[4] = 2u | (2u << 16);          // tile_dim1=2, tile_dim2=2
        g1[5] = (uint32_t)HN;             // tensor_dim0_stride=128
        g1[6] = (uint32_t)(HN * HN) << 16;// tensor_dim1_stride[15:0]=16384
        g1[7] = 0u;                       // tensor_dim1_stride[47:16]=0
        v4u g2;
        g2[0] = (uint32_t)HN;             // tensor_dim2=128
        g2[1] = 0u; g2[2] = 0u; g2[3] = 0u;
        v4u g3; g3[0] = 0u; g3[1] = 0u; g3[2] = 0u; g3[3] = 0u;
        asm volatile("tensor_load_to_lds %0, %1, %2, %3"
                     :: "s"(g0), "s"(g1), "s"(g2), "s"(g3) : "memory");

        // label row: CDNA5 async global->LDS (32 lanes x b128 = 512B row)
        const uint32_t byteOff = threadIdx.x * 16u;
        const char* gp = (const char*)(lb + ((size_t)hn * HN + wn) * HN) + byteOff;
        uint32_t ldsAddr = (uint32_t)(uintptr_t)(&lrow[0]) + byteOff;
        asm volatile("global_load_async_to_lds_b128 %0, %1, off"
                     :: "v"(ldsAddr), "v"(gp) : "memory");

        __builtin_amdgcn_s_wait_tensorcnt(0);
        asm volatile("s_wait_asynccnt 0x0" ::: "memory");
    }
    __syncthreads();

    // ---- per-lane D axis ----
    const int d = (int)threadIdx.x;
    const int ed = fd ? (HN - 1 - d) : d;
    const float sd = (ed - HC) * invz + HC;
    const int d0i = (int)floorf(sd);
    const float td = sd - (float)d0i;
    const float wd0 = (1.0f - td) * ((d0i     >= 0 && d0i     < HN) ? 1.0f : 0.0f);
    const float wd1 = td          * ((d0i + 1 >= 0 && d0i + 1 < HN) ? 1.0f : 0.0f);
    const int d0 = min(max(d0i, 0), HN - 1), d1 = min(max(d0i + 1, 0), HN - 1);

    float r0 = rows[0][d0] * wd0 + rows[0][d1] * wd1;   // (ho,   wo  )
    float r1 = rows[1][d0] * wd0 + rows[1][d1] * wd1;   // (ho,   wo+1)
    float r2 = rows[2][d0] * wd0 + rows[2][d1] * wd1;   // (ho+1, wo  )
    float r3 = rows[3][d0] * wd0 + rows[3][d1] * wd1;   // (ho+1, wo+1)
    float v = wsh[0] * (wsw[0] * r0 + wsw[1] * r1) + wsh[1] * (wsw[0] * r2 + wsw[1] * r3);

    // gaussian noise (Box-Muller on counter RNG); stdev is block-uniform -> scalar branch
    if (stdev > 0.0f) {
        uint32_t vox = (uint32_t)(((h * HN) + w) * HN + d);
        U2 rb = threefry2x32(kn0, kn1, vox, 0x9E3779B9u);
        float u1 = u01(rb.x) + 1.0e-12f;
        float u2 = u01(rb.y);
        float mag = sqrtf(-2.0f * __logf(u1));
        v += stdev * mag * __cosf(6.28318530718f * u2);
    }

    const size_t oOff = sampOff + ((size_t)h * HN + w) * HN + d;
    oimg[oOff] = v;   // pre-gamma image (gamma pass re-reads from L2)

    // label: nearest in all 3 axes, zero padding
    const int dni = (int)rintf(sd);
    const bool dv = (dni >= 0) && (dni < HN);
    const int dn = min(max(dni, 0), HN - 1);
    olbl[oOff] = (hv && wv && dv) ? lrow[dn] : 0.0f;

    // ---- block min/max reduction -> native float global atomics ----
    smin[d] = v;
    smax[d] = v;
    __syncthreads();
    for (int s = 64; s > 0; s >>= 1) {
        if (d < s) {
            smin[d] = fminf(smin[d], smin[d + s]);
            smax[d] = fmaxf(smax[d], smax[d + s]);
        }
        __syncthreads();
    }
    if (d == 0) {
        __hip_atomic_fetch_min(&minmax[2 * b],     smin[0], __ATOMIC_RELAXED, __HIP_MEMORY_SCOPE_AGENT);
        __hip_atomic_fetch_max(&minmax[2 * b + 1], smax[0], __ATOMIC_RELAXED, __HIP_MEMORY_SCOPE_AGENT);
    }
}

// ---------------- kernel 3: gamma contrast in place, float4 streaming (L2-resident) ----------------
__global__ void aug_gamma_kernel(float4* __restrict__ oimg, const float* __restrict__ params,
                                 const float* __restrict__ minmax, int nvec) {
    int i = (int)(blockIdx.x * blockDim.x + threadIdx.x);
    if (i >= nvec) return;
    int b = i >> 19;                                  // (i*4) >> 21 : 128^3 voxels per sample
    float gamma = params[b * 8 + 5];
    float mn = minmax[2 * b], mx = minmax[2 * b + 1];
    float rng = mx - mn;
    float inv = 1.0f / (rng + 1e-7f);
    float4 v = oimg[i];
    float* e = &v.x;
#pragma unroll
    for (int k = 0; k < 4; ++k) {
        float xn = fminf(fmaxf((e[k] - mn) * inv, 1e-6f), 1.0f);
        e[k] = exp2f(gamma * log2f(xn)) * rng + mn;   // xn^gamma (applied even when gamma==1)
    }
    oimg[i] = v;
}

extern "C" void kernel_launch(void* const* d_in, const int* in_sizes, int n_in,
                              void* d_out, int out_size, void* d_ws, size_t ws_size,
                              hipStream_t stream) {
    const float* img = (const float*)d_in[0];
    const float* lbl = (const float*)d_in[1];
    const int vox = HN * HN * HN;
    const int B = in_sizes[0] / vox;          // C == 1

    float* oimg = (float*)d_out;
    float* olbl = oimg + (size_t)B * vox;

    float* params = (float*)d_ws;             // 8 floats / sample
    float* minmax = params + 8 * B;           // 2 floats / sample

    aug_params_kernel<<<1, 32, 0, stream>>>(params, minmax, B);
    aug_line_kernel<<<B * HN * HN, HN, 0, stream>>>(img, lbl, oimg, olbl, params, minmax);
    int nvec = (B * vox) / 4;
    aug_gamma_kernel<<<(nvec + 255) / 256, 256, 0, stream>>>((float4*)oimg, params, minmax, nvec);
}